// DiffAllocator_26268019982505
// MI455X (gfx1250) — compile-verified
//
#include <hip/hip_runtime.h>
#include <math.h>

// ---------------------------------------------------------------------------
// DiffAllocator Sinkhorn (L=64, B=7, 200 iters) for MI455X / gfx1250.
// Latency-bound: single wave32, all state in VGPRs.
//  - whole computation in log2 domain -> raw v_exp_f32 / v_log_f32
//  - 32-lane reductions entirely in VALU: 4 DPP16 stages + v_permlanex16
//    cross-half exchange (zero LDS round-trips in the iteration loop)
//  - V_WMMA_F32_16X16X4_F32 for the final global normalization sum
// ---------------------------------------------------------------------------

namespace {
constexpr int   kB     = 7;
constexpr int   kIters = 200;
constexpr float kEps   = 0.02f;
constexpr float kTiny  = 1e-40f;
constexpr float kRLn2  = 1.4426950408889634f;  // 1/ln(2)
}

typedef __attribute__((ext_vector_type(2))) float v2f;
typedef __attribute__((ext_vector_type(8))) float v8f;

#if defined(__has_builtin)
#  if __has_builtin(__builtin_amdgcn_wmma_f32_16x16x4_f32)
#    define DA_HAVE_WMMA 1
#  endif
#  if __has_builtin(__builtin_amdgcn_ds_swizzle)
#    define DA_HAVE_SWIZZLE 1
#  endif
#  if __has_builtin(__builtin_amdgcn_exp2f) && __has_builtin(__builtin_amdgcn_logf)
#    define DA_HAVE_TRANS 1
#  endif
#  if __has_builtin(__builtin_amdgcn_update_dpp)
#    define DA_HAVE_DPP 1
#  endif
#  if __has_builtin(__builtin_amdgcn_permlanex16)
#    define DA_HAVE_PERMX16 1
#  endif
#  if __has_builtin(__builtin_amdgcn_rcpf)
#    define DA_HAVE_RCP 1
#  endif
#endif
#ifndef DA_HAVE_WMMA
#  define DA_HAVE_WMMA 0
#endif
#ifndef DA_HAVE_SWIZZLE
#  define DA_HAVE_SWIZZLE 0
#endif
#ifndef DA_HAVE_TRANS
#  define DA_HAVE_TRANS 0
#endif
#ifndef DA_HAVE_DPP
#  define DA_HAVE_DPP 0
#endif
#ifndef DA_HAVE_PERMX16
#  define DA_HAVE_PERMX16 0
#endif
#ifndef DA_HAVE_RCP
#  define DA_HAVE_RCP 0
#endif

// Raw hardware transcendentals: v_exp_f32 is exp2, v_log_f32 is log2.
__device__ __forceinline__ float fexp2(float x) {
#if DA_HAVE_TRANS
  return __builtin_amdgcn_exp2f(x);
#else
  return exp2f(x);
#endif
}
__device__ __forceinline__ float flog2(float x) {
#if DA_HAVE_TRANS
  return __builtin_amdgcn_logf(x);
#else
  return log2f(x);
#endif
}

// DPP16 lane-pairing permutes (pure VALU). Each is an involution pairing
// disjoint partners at its level, so the 5-stage sequence
// quad_xor1 -> quad_xor2 -> row_half_mirror -> row_mirror -> cross-half
// yields a full 32-lane reduction broadcast to every lane.
enum {
  DPP_QUAD_XOR1       = 0xB1,   // quad_perm [1,0,3,2]
  DPP_QUAD_XOR2       = 0x4E,   // quad_perm [2,3,0,1]
  DPP_ROW_HALF_MIRROR = 0x141,  // l <-> 7-l within each 8
  DPP_ROW_MIRROR      = 0x140   // l <-> 15-l within each 16
};

template <int CTRL>
__device__ __forceinline__ float dpp_perm(float v) {
  int s = __float_as_int(v);
  return __int_as_float(__builtin_amdgcn_update_dpp(s, s, CTRL, 0xF, 0xF, false));
}

// Cross-half (row0 <-> row1) exchange. Preferred: v_permlanex16_b32 with the
// identity lane-select (each lane reads lane (l%16) of the opposite row) --
// pure VALU, no LDS. Fallbacks: ds_swizzle SWAP16, then __shfl_xor.
__device__ __forceinline__ float xhalf(float v) {
#if DA_HAVE_PERMX16
  int s = __float_as_int(v);
  return __int_as_float(__builtin_amdgcn_permlanex16(
      s, s, 0x76543210, 0xFEDCBA98, /*fi=*/false, /*bound_ctrl=*/false));
#elif DA_HAVE_SWIZZLE
  return __int_as_float(
      __builtin_amdgcn_ds_swizzle(__float_as_int(v), (16 << 10) | 0x1F));
#else
  return __shfl_xor(v, 16, 32);
#endif
}
// Generic XOR exchange (fallback path when DPP unavailable).
template <int XORM>
__device__ __forceinline__ float swz_xor(float v) {
#if DA_HAVE_SWIZZLE
  return __int_as_float(
      __builtin_amdgcn_ds_swizzle(__float_as_int(v), (XORM << 10) | 0x1F));
#else
  return __shfl_xor(v, XORM, 32);
#endif
}

// Batched reduction stages over all 7 columns (issue all exchanges first).
#if DA_HAVE_DPP
template <int CTRL>
__device__ __forceinline__ void stage_dpp_max(float (&v)[kB]) {
  float t[kB];
#pragma unroll
  for (int j = 0; j < kB; ++j) t[j] = dpp_perm<CTRL>(v[j]);
#pragma unroll
  for (int j = 0; j < kB; ++j) v[j] = fmaxf(v[j], t[j]);
}
template <int CTRL>
__device__ __forceinline__ void stage_dpp_sum(float (&v)[kB]) {
  float t[kB];
#pragma unroll
  for (int j = 0; j < kB; ++j) t[j] = dpp_perm<CTRL>(v[j]);
#pragma unroll
  for (int j = 0; j < kB; ++j) v[j] += t[j];
}
#endif
__device__ __forceinline__ void stage_xhalf_max(float (&v)[kB]) {
  float t[kB];
#pragma unroll
  for (int j = 0; j < kB; ++j) t[j] = xhalf(v[j]);
#pragma unroll
  for (int j = 0; j < kB; ++j) v[j] = fmaxf(v[j], t[j]);
}
__device__ __forceinline__ void stage_xhalf_sum(float (&v)[kB]) {
  float t[kB];
#pragma unroll
  for (int j = 0; j < kB; ++j) t[j] = xhalf(v[j]);
#pragma unroll
  for (int j = 0; j < kB; ++j) v[j] += t[j];
}
template <int M>
__device__ __forceinline__ void stage_swz_max(float (&v)[kB]) {
  float t[kB];
#pragma unroll
  for (int j = 0; j < kB; ++j) t[j] = swz_xor<M>(v[j]);
#pragma unroll
  for (int j = 0; j < kB; ++j) v[j] = fmaxf(v[j], t[j]);
}
template <int M>
__device__ __forceinline__ void stage_swz_sum(float (&v)[kB]) {
  float t[kB];
#pragma unroll
  for (int j = 0; j < kB; ++j) t[j] = swz_xor<M>(v[j]);
#pragma unroll
  for (int j = 0; j < kB; ++j) v[j] += t[j];
}

// Full 32-lane reductions over the 7-column state (result in every lane).
__device__ __forceinline__ void red_max(float (&v)[kB]) {
#if DA_HAVE_DPP
  stage_dpp_max<DPP_QUAD_XOR1>(v);
  stage_dpp_max<DPP_QUAD_XOR2>(v);
  stage_dpp_max<DPP_ROW_HALF_MIRROR>(v);
  stage_dpp_max<DPP_ROW_MIRROR>(v);
  stage_xhalf_max(v);
#else
  stage_swz_max<1>(v); stage_swz_max<2>(v); stage_swz_max<4>(v);
  stage_swz_max<8>(v); stage_xhalf_max(v);
#endif
}
__device__ __forceinline__ void red_sum(float (&v)[kB]) {
#if DA_HAVE_DPP
  stage_dpp_sum<DPP_QUAD_XOR1>(v);
  stage_dpp_sum<DPP_QUAD_XOR2>(v);
  stage_dpp_sum<DPP_ROW_HALF_MIRROR>(v);
  stage_dpp_sum<DPP_ROW_MIRROR>(v);
  stage_xhalf_sum(v);
#else
  stage_swz_sum<1>(v); stage_swz_sum<2>(v); stage_swz_sum<4>(v);
  stage_swz_sum<8>(v); stage_xhalf_sum(v);
#endif
}

// Scalar 32-lane sum (prologue).
__device__ __forceinline__ float wave_sum32(float v) {
#if DA_HAVE_DPP
  v += dpp_perm<DPP_QUAD_XOR1>(v);
  v += dpp_perm<DPP_QUAD_XOR2>(v);
  v += dpp_perm<DPP_ROW_HALF_MIRROR>(v);
  v += dpp_perm<DPP_ROW_MIRROR>(v);
#else
  v += swz_xor<1>(v); v += swz_xor<2>(v);
  v += swz_xor<4>(v); v += swz_xor<8>(v);
#endif
  v += xhalf(v);
  return v;
}

// log-depth trees over 7 values (FP not reassociated by default -> explicit)
__device__ __forceinline__ float max7(const float* x) {
  float a = fmaxf(x[0], x[1]);
  float b = fmaxf(x[2], x[3]);
  float c = fmaxf(x[4], x[5]);
  return fmaxf(fmaxf(a, b), fmaxf(c, x[6]));
}
__device__ __forceinline__ float sum7(const float* x) {
  float a = x[0] + x[1];
  float b = x[2] + x[3];
  float c = x[4] + x[5];
  return (a + b) + (c + x[6]);
}

__global__ __launch_bounds__(32, 1)
void diffalloc_sinkhorn(const float* __restrict__ theta,
                        const float* __restrict__ phi,
                        const float* __restrict__ nbuf,
                        const float* __restrict__ sens,
                        const float* __restrict__ err,
                        float* __restrict__ out) {
  const int lane = (int)threadIdx.x;   // 0..31, wave32
  const int r0 = lane;                 // lane owns rows r0 and r1
  const int r1 = lane + 32;

  // ---- load tiny inputs; everything register-resident afterwards ----
  float errv[kB], phi2[kB];
#pragma unroll
  for (int j = 0; j < kB; ++j) {
    errv[j] = err[j];
    phi2[j] = phi[j] * kRLn2;          // log2 domain
  }
  const float n0 = nbuf[r0], n1 = nbuf[r1];
  const float s0 = sens[r0], s1 = sens[r1];

  float th0[kB], th1[kB];
#pragma unroll
  for (int j = 0; j < kB; ++j) {
    th0[j] = theta[r0 * kB + j];
    th1[j] = theta[r1 * kB + j];
  }

  // ---- row marginal: a = n / sum(n) ----
  const float nsum = wave_sum32(n0 + n1);
  const float log_a0 = flog2(n0 / nsum + kTiny);   // log2(a + TINY)
  const float log_a1 = flog2(n1 / nsum + kTiny);

  // ---- col marginal: b = softmax(phi); log2(b + TINY), replicated/lane ----
  const float pmax = max7(phi2);
  float pt[kB];
#pragma unroll
  for (int j = 0; j < kB; ++j) pt[j] = fexp2(phi2[j] - pmax);
  const float lse_phi2 = pmax + flog2(sum7(pt));
  float log_b[kB];
#pragma unroll
  for (int j = 0; j < kB; ++j)
    log_b[j] = flog2(fexp2(phi2[j] - lse_phi2) + kTiny);

  // ---- log2-kernel K2 = (theta - n*sens*err) / (EPS*ln2) ----
  const float scale = kRLn2 / kEps;
  const float c0 = n0 * s0, c1 = n1 * s1;
  float K0[kB], K1[kB];
#pragma unroll
  for (int j = 0; j < kB; ++j) {
    K0[j] = (th0[j] - c0 * errv[j]) * scale;
    K1[j] = (th1[j] - c1 * errv[j]) * scale;
  }

  // ---- Sinkhorn iterations (all quantities = reference values / ln2) ----
  float f0 = 0.f, f1 = 0.f;
  float g[kB];
#pragma unroll
  for (int j = 0; j < kB; ++j) g[j] = 0.f;

  for (int it = 0; it < kIters; ++it) {
    // f-step: f_i = log_a_i - lse2_j(K2_ij + g_j)   (pure in-lane)
    float y0[kB], y1[kB];
#pragma unroll
    for (int j = 0; j < kB; ++j) {
      y0[j] = K0[j] + g[j];
      y1[j] = K1[j] + g[j];
    }
    const float a0 = max7(y0), a1 = max7(y1);
    float e0[kB], e1[kB];
#pragma unroll
    for (int j = 0; j < kB; ++j) {
      e0[j] = fexp2(y0[j] - a0);
      e1[j] = fexp2(y1[j] - a1);
    }
    f0 = log_a0 - (a0 + flog2(sum7(e0)));
    f1 = log_a1 - (a1 + flog2(sum7(e1)));

    // g-step: g_j = log_b_j - lse2_i(K2_ij + f_i)
    // 7 independent 32-lane reductions, all VALU (DPP + permlanex16).
    float x0[kB], x1[kB], mj[kB], sj[kB];
#pragma unroll
    for (int j = 0; j < kB; ++j) {
      x0[j] = K0[j] + f0;
      x1[j] = K1[j] + f1;
      mj[j] = fmaxf(x0[j], x1[j]);
    }
    red_max(mj);
#pragma unroll
    for (int j = 0; j < kB; ++j)
      sj[j] = fexp2(x0[j] - mj[j]) + fexp2(x1[j] - mj[j]);
    red_sum(sj);
#pragma unroll
    for (int j = 0; j < kB; ++j)
      g[j] = log_b[j] - (mj[j] + flog2(sj[j]));
  }

  // ---- P = 2^(K2 + f2 + g2) = exp(K + f + g); normalize by global sum ----
  float p0[kB], p1[kB], q[kB];
#pragma unroll
  for (int j = 0; j < kB; ++j) {
    p0[j] = fexp2(K0[j] + f0 + g[j]);
    p1[j] = fexp2(K1[j] + f1 + g[j]);
    q[j]  = p0[j] + p1[j];
  }
  const float psum = sum7(q);

  float total;
#if DA_HAVE_WMMA
  {
    // 32-lane sum via V_WMMA_F32_16X16X4_F32 with B = ones:
    //   A layout (ISA 7.12.2): lane<16 -> {A(l,0),A(l,1)}, lane>=16 ->
    //   {A(l-16,2),A(l-16,3)}.  A(m,0)=ps_m, A(m,2)=ps_{m+16}, rest 0.
    //   D(m,n) = ps_m + ps_{m+16} for all n.
    // C/D layout: lanes<16 hold M=0..7, lanes>=16 hold M=8..15, so the
    // in-lane 8-VGPR sum gives one half per lane-half; one cross-half
    // exchange completes the exact 32-lane total, broadcast to all lanes.
    v2f amat; amat.x = psum; amat.y = 0.0f;
    v2f bone; bone.x = 1.0f; bone.y = 1.0f;
    v8f cz = {};
    v8f d = __builtin_amdgcn_wmma_f32_16x16x4_f32(
        /*neg_a=*/false, amat, /*neg_b=*/false, bone,
        /*c_mod=*/(short)0, cz, /*reuse_a=*/false, /*reuse_b=*/false);
    float h = ((d[0] + d[1]) + (d[2] + d[3])) + ((d[4] + d[5]) + (d[6] + d[7]));
    total = h + xhalf(h);
  }
#else
  total = wave_sum32(psum);
#endif

  // Normalization constant: raw v_rcp_f32 is ample for a scale factor.
#if DA_HAVE_RCP
  const float inv = __builtin_amdgcn_rcpf(total + kTiny);
#else
  const float inv = 1.0f / (total + kTiny);
#endif
#pragma unroll
  for (int j = 0; j < kB; ++j) {
    out[r0 * kB + j] = p0[j] * inv;
    out[r1 * kB + j] = p1[j] * inv;
  }
}

extern "C" void kernel_launch(void* const* d_in, const int* in_sizes, int n_in,
                              void* d_out, int out_size, void* d_ws, size_t ws_size,
                              hipStream_t stream) {
  (void)in_sizes; (void)n_in; (void)out_size; (void)d_ws; (void)ws_size;
  const float* theta = (const float*)d_in[0];  // [64,7]
  const float* phi   = (const float*)d_in[1];  // [7]
  const float* n     = (const float*)d_in[2];  // [64]
  const float* sens  = (const float*)d_in[3];  // [64]
  const float* err   = (const float*)d_in[4];  // [7]
  float* out = (float*)d_out;                  // [64,7]

  // Single wave32: 200 serially-dependent global reductions -> one wave with
  // register-resident state minimizes per-iteration latency.
  diffalloc_sinkhorn<<<dim3(1), dim3(32), 0, stream>>>(theta, phi, n, sens, err, out);
}